// AdaptiveGCNLayer_64183991272032
// MI455X (gfx1250) — compile-verified
//
#include <hip/hip_runtime.h>

typedef __attribute__((ext_vector_type(2))) float v2f;
typedef __attribute__((ext_vector_type(8))) float v8f;

#define TILE 32
#define KC   32
#define PAD  36   // floats per LDS row: 144B rows stay 16B-aligned for async b128

enum { EPI_STORE = 0, EPI_MASK = 1, EPI_FINAL = 2 };

// gfx1250 async global->LDS copy, 16B per lane, tracked by ASYNCcnt.
__device__ __forceinline__ void async_b128(unsigned ldsOff, const float* g) {
    asm volatile("global_load_async_to_lds_b128 %0, %1, off"
                 :: "v"(ldsOff), "v"(g) : "memory");
}
__device__ __forceinline__ void wait_async0() {
    asm volatile("s_wait_asynccnt 0x0" ::: "memory");
}

// C[M,N] = A[M,K] @ op(B) ;  BT=true means B is stored [N][K] (i.e. A @ B^T)
template<int EPI, bool BT>
__launch_bounds__(128)
__global__ void gemm32_wmma(const float* __restrict__ A, const float* __restrict__ B,
                            float* __restrict__ C,
                            int M, int N, int K, int lda, int ldb, int ldc,
                            const float* __restrict__ dvec,   // EPI_MASK: d ; EPI_FINAL: dis
                            float* __restrict__ degOut,       // EPI_MASK only
                            const float* __restrict__ bias)   // EPI_FINAL only
{
    // double-buffered tiles: [buf][A=0/B=1][TILE*PAD]
    __shared__ float smem[2][2][TILE * PAD];

    const int tid  = threadIdx.x;
    const int lane = tid & 31;
    const int wave = tid >> 5;
    const int wm   = (wave >> 1) * 16;   // wave tile row offset within block tile
    const int wn   = (wave & 1) * 16;    // wave tile col offset
    const int half = lane >> 4;          // 0: lanes 0-15 (K=0,1), 1: lanes 16-31 (K=2,3)
    const int l15  = lane & 15;

    const int rowBase = blockIdx.y * TILE;
    const int colBase = blockIdx.x * TILE;

    const int r2 = tid >> 3;            // 0..15 : tile row handled in first pass
    const int c4 = (tid & 7) << 2;      // 0,4,...,28 : 16B chunk within row

    auto issueTile = [&](int k0, int buf) {
        const unsigned bA = (unsigned)(uintptr_t)&smem[buf][0][0];
        const unsigned bB = (unsigned)(uintptr_t)&smem[buf][1][0];
        // A tile: rows rowBase..+31, cols k0..+31 -> sA[r][k]
        async_b128(bA + (unsigned)(r2 * PAD + c4) * 4u,
                   A + (size_t)(rowBase + r2) * lda + k0 + c4);
        async_b128(bA + (unsigned)((r2 + 16) * PAD + c4) * 4u,
                   A + (size_t)(rowBase + r2 + 16) * lda + k0 + c4);
        if (BT) {
            // B rows are n (A@B^T): B[n][k] -> sB[n][k]
            async_b128(bB + (unsigned)(r2 * PAD + c4) * 4u,
                       B + (size_t)(colBase + r2) * ldb + k0 + c4);
            async_b128(bB + (unsigned)((r2 + 16) * PAD + c4) * 4u,
                       B + (size_t)(colBase + r2 + 16) * ldb + k0 + c4);
        } else {
            // B rows are k: B[k][n] -> sB[k][n] (natural order, contiguous b128)
            async_b128(bB + (unsigned)(r2 * PAD + c4) * 4u,
                       B + (size_t)(k0 + r2) * ldb + colBase + c4);
            async_b128(bB + (unsigned)((r2 + 16) * PAD + c4) * 4u,
                       B + (size_t)(k0 + r2 + 16) * ldb + colBase + c4);
        }
    };

    v8f acc = {};
    const int nT = K / KC;
    int buf = 0;
    issueTile(0, 0);

    for (int t = 0; t < nT; ++t) {
        wait_async0();        // own wave's async writes to smem[buf] complete
        __syncthreads();      // -> all waves' writes visible; prev reads done
        if (t + 1 < nT) issueTile((t + 1) * KC, buf ^ 1);   // overlap with compute

        const float* sA = smem[buf][0];
        const float* sB = smem[buf][1];
        #pragma unroll
        for (int kk = 0; kk < KC; kk += 4) {
            const int ka = kk + half * 2;
            v2f aF, bF;
            aF.x = sA[(wm + l15) * PAD + ka];
            aF.y = sA[(wm + l15) * PAD + ka + 1];
            if (BT) {
                bF.x = sB[(wn + l15) * PAD + ka];
                bF.y = sB[(wn + l15) * PAD + ka + 1];
            } else {
                bF.x = sB[ka * PAD + wn + l15];
                bF.y = sB[(ka + 1) * PAD + wn + l15];
            }
            acc = __builtin_amdgcn_wmma_f32_16x16x4_f32(false, aF, false, bF,
                                                        (short)0, acc, false, false);
        }
        buf ^= 1;
    }

    // ---- epilogue: C/D layout -> lanes 0-15: M=r, lanes 16-31: M=r+8 ----
    #pragma unroll
    for (int r = 0; r < 8; ++r) {
        const int gm = rowBase + wm + r + half * 8;
        const int gn = colBase + wn + l15;
        const unsigned ci = (unsigned)gm * (unsigned)ldc + (unsigned)gn;
        const float v = acc[r];
        if (EPI == EPI_STORE) {
            C[ci] = v;
        } else if (EPI == EPI_FINAL) {
            C[ci] = dvec[gm] * v + bias[gn];
        } else {  // EPI_MASK
            const float adjv = v + ((gm == gn) ? 1.0f : 0.0f);
            const float mval = dvec[gm] * adjv * dvec[gn];   // nan/inf propagate
            const float a = (gm == gn) ? 1.0f : ((mval != 0.0f) ? 1.0f : 0.0f);
            C[ci] = a;
            float s = a;                  // reduce across 16-lane half (mask row)
            s += __shfl_xor(s, 1);
            s += __shfl_xor(s, 2);
            s += __shfl_xor(s, 4);
            s += __shfl_xor(s, 8);
            if (l15 == 0) atomicAdd(&degOut[gm], s);
        }
    }
}

__global__ void zero_f32(float* __restrict__ p, int n) {
    int i = blockIdx.x * blockDim.x + threadIdx.x;
    if (i < n) p[i] = 0.0f;
}

// xsum[c] = sum_i x[i][c]  (grid.x covers cols, grid.y covers row chunks of 256)
__global__ void colsum_kernel(const float* __restrict__ x, float* __restrict__ xsum,
                              int rows, int cols) {
    const int c  = blockIdx.x * blockDim.x + threadIdx.x;
    const int r0 = blockIdx.y * 256;
    float s = 0.0f;
    for (int r = r0; r < r0 + 256; ++r) s += x[(size_t)r * cols + c];
    atomicAdd(&xsum[c], s);
}

// d[i] = (S[i,:] . xsum + 1)^(-1/2)   (rowsum(adj) via column-sum identity)
__global__ void rowdot_rsqrt(const float* __restrict__ S, const float* __restrict__ xsum,
                             float* __restrict__ d, int rows, int cols) {
    const int w    = (blockIdx.x * blockDim.x + threadIdx.x) >> 5;
    const int lane = threadIdx.x & 31;
    if (w >= rows) return;
    const float* sr = S + (size_t)w * cols;
    float acc = 0.0f;
    for (int c = lane; c < cols; c += 32) acc += sr[c] * xsum[c];
    for (int off = 16; off; off >>= 1) acc += __shfl_xor(acc, off);
    if (lane == 0) d[w] = rsqrtf(acc + 1.0f);   // neg -> nan, 0 -> inf (matches **-0.5)
}

__global__ void deg_to_dis(const float* __restrict__ deg, float* __restrict__ dis, int n) {
    int i = blockIdx.x * blockDim.x + threadIdx.x;
    if (i < n) {
        float g = deg[i];
        dis[i] = (g > 0.0f) ? rsqrtf(g) : 0.0f;
    }
}

// Y[j][c] *= dis[j]   (in place)
__global__ void scale_rows(float* __restrict__ Y, const float* __restrict__ dis,
                           int rows, int cols) {
    int i = blockIdx.x * blockDim.x + threadIdx.x;
    if (i < rows * cols) Y[i] *= dis[i / cols];
}

extern "C" void kernel_launch(void* const* d_in, const int* in_sizes, int n_in,
                              void* d_out, int out_size, void* d_ws, size_t ws_size,
                              hipStream_t stream) {
    const float* x  = (const float*)d_in[0];   // [4096, 512]
    const float* Wa = (const float*)d_in[1];   // [512, 512]
    const float* Wg = (const float*)d_in[2];   // [512, 512]
    const float* bg = (const float*)d_in[3];   // [512]
    float* out = (float*)d_out;                // [4096, 512]

    const int N = 4096, C = 512;
    float* ws   = (float*)d_ws;
    float* S    = ws;                          // N*C  : x @ W_adj
    float* Y    = S  + (size_t)N * C;          // N*C  : x @ W_gcn (dis_j-scaled in place)
    float* Am   = Y  + (size_t)N * C;          // N*N  : 0/1 mask
    float* xsum = Am + (size_t)N * N;          // C
    float* dr   = xsum + C;                    // N : rowsum(adj)^-1/2
    float* deg  = dr + N;                      // N
    float* dis  = deg + N;                     // N

    // zero xsum + dr + deg + dis region
    const int zn = C + 3 * N;
    zero_f32<<<(zn + 255) / 256, 256, 0, stream>>>(xsum, zn);

    const dim3 blk(128);
    // S = x @ W_adj ; Y = x @ W_gcn
    gemm32_wmma<EPI_STORE, false><<<dim3(C / TILE, N / TILE), blk, 0, stream>>>(
        x, Wa, S, N, C, C, C, C, C, nullptr, nullptr, nullptr);
    gemm32_wmma<EPI_STORE, false><<<dim3(C / TILE, N / TILE), blk, 0, stream>>>(
        x, Wg, Y, N, C, C, C, C, C, nullptr, nullptr, nullptr);

    // degree vector of adaptive adjacency via column-sum identity
    colsum_kernel<<<dim3(C / 256, N / 256), 256, 0, stream>>>(x, xsum, N, C);
    rowdot_rsqrt<<<N / 4, 128, 0, stream>>>(S, xsum, dr, N, C);

    // mask pass: adj tile = S @ x^T (+I), scale by d, compare != 0, accumulate deg
    gemm32_wmma<EPI_MASK, true><<<dim3(N / TILE, N / TILE), blk, 0, stream>>>(
        S, x, Am, N, N, C, C, C, N, dr, deg, nullptr);

    deg_to_dis<<<(N + 255) / 256, 256, 0, stream>>>(deg, dis, N);
    scale_rows<<<((size_t)N * C + 255) / 256, 256, 0, stream>>>(Y, dis, N, C);

    // out = dis_i * (Am @ Y_scaled) + bias
    gemm32_wmma<EPI_FINAL, false><<<dim3(C / TILE, N / TILE), blk, 0, stream>>>(
        Am, Y, out, N, C, N, N, C, C, dis, nullptr, bg);
}